// GCN_15917148799233
// MI455X (gfx1250) — compile-verified
//
#include <hip/hip_runtime.h>
#include <math.h>
#include <stdint.h>

typedef __attribute__((ext_vector_type(2))) float v2f;
typedef __attribute__((ext_vector_type(8))) float v8f;

#define NNODES 100000
#define FIN    512
#define HID    256
#define NCLS   40

// ---------------------------------------------------------------------------
// CDNA5 async global->LDS copies (ASYNCcnt-tracked DMA, no VGPR round trip).
// LDS byte address = low 32 bits of the flat pointer.
// Completed with s_wait_asynccnt 0 before the workgroup barrier.
// ---------------------------------------------------------------------------
__device__ __forceinline__ void async_g2l_b128(const float* g, const float* l) {
    const uint32_t lds = (uint32_t)(uintptr_t)l;
    asm volatile("global_load_async_to_lds_b128 %0, %1, off"
                 :: "v"(lds), "v"(g) : "memory");
}
__device__ __forceinline__ void async_g2l_b32(const float* g, const float* l) {
    const uint32_t lds = (uint32_t)(uintptr_t)l;
    asm volatile("global_load_async_to_lds_b32 %0, %1, off"
                 :: "v"(lds), "v"(g) : "memory");
}
__device__ __forceinline__ void wait_async0() {
    asm volatile("s_wait_asynccnt 0x0" ::: "memory");
}

// ---------------------------------------------------------------------------
// Degree accumulation: deg[col[e]] += 1  (self-loop handled as +1 in rsqrt)
// ---------------------------------------------------------------------------
__global__ void gcn_degree(const int* __restrict__ col, float* __restrict__ deg, int nE) {
    for (int i = blockIdx.x * blockDim.x + threadIdx.x; i < nE;
         i += gridDim.x * blockDim.x) {
        atomicAdd(&deg[col[i]], 1.0f);
    }
}

__global__ void gcn_rsqrt(float* __restrict__ dinv, int n) {
    int i = blockIdx.x * blockDim.x + threadIdx.x;
    if (i < n) {
        float d = dinv[i] + 1.0f;          // +1 self loop; always > 0
        dinv[i] = rsqrtf(d);
    }
}

// ---------------------------------------------------------------------------
// FP32 WMMA GEMM:  C[M x NOUT] = A[M x KDIM] @ B[KDIM x NOUT]
// Block = 128 threads (4 waves). Block tile = 16 rows x 64 cols; each wave
// owns one 16x16 tile (V_WMMA_F32_16X16X4_F32), K-chunked by 32.
// A is staged transposed [k][m] and B staged [k][n] in LDS via async DMA,
// so the inner loop is branch-free ds_load + v_wmma only.
// grid = (M/16, NOUT tiles of 64); cols >= NOUT are zero-padded in LDS.
// ---------------------------------------------------------------------------
template <int KDIM, int NOUT>
__global__ void gcn_gemm_wmma(const float* __restrict__ A,
                              const float* __restrict__ B,
                              float* __restrict__ Cout) {
    __shared__ float As[32 * 16];          // [k][m] chunk, 2 KB
    __shared__ float Bs[32 * 64];          // [k][n-in-block] chunk, 8 KB

    const int tid  = threadIdx.x;
    const int lane = tid & 31;
    const int wave = tid >> 5;
    const int half = lane >> 4;            // 0: lanes 0-15, 1: lanes 16-31
    const int lm   = lane & 15;
    const int m0   = blockIdx.x * 16;
    const int n0b  = blockIdx.y * 64;
    const int cb   = wave * 16 + lm;       // column within block tile (0..63)
    const int col  = n0b + cb;

    // Zero-pad Bs once when the 64-wide tile overhangs NOUT (GEMM2 case).
    if constexpr ((NOUT & 63) != 0) {
        for (int i = tid; i < 32 * 64; i += 128) Bs[i] = 0.0f;
        __syncthreads();
    }

    // Per-thread staging roles (fixed across chunks):
    const int br  = tid >> 2;              // B: row 0..31 of chunk
    const int bc  = (tid & 3) << 4;        // B: 16-col segment start in tile
    const int ar  = tid >> 3;              // A: row 0..15
    const int ac  = (tid & 7) << 2;        // A: 4-col segment start in chunk

    v8f acc = {};

    for (int k0 = 0; k0 < KDIM; k0 += 32) {
        // ---- async-stage B chunk: rows k0..k0+31, cols n0b..n0b+63 ----
        {
            const float* gsrc = B + (size_t)(k0 + br) * NOUT + n0b + bc;
            float*       ldst = &Bs[br * 64 + bc];
            if (n0b + bc + 16 <= NOUT) {
#pragma unroll
                for (int q = 0; q < 16; q += 4) async_g2l_b128(gsrc + q, ldst + q);
            } else {
#pragma unroll
                for (int q = 0; q < 16; ++q)
                    if (n0b + bc + q < NOUT) async_g2l_b32(gsrc + q, ldst + q);
            }
        }
        // ---- async-stage A chunk transposed into [k][m] ----
        {
            const float* gsrc = A + (size_t)(m0 + ar) * KDIM + k0 + ac;
#pragma unroll
            for (int i = 0; i < 4; ++i)
                async_g2l_b32(gsrc + i, &As[(ac + i) * 16 + ar]);
        }
        wait_async0();
        __syncthreads();

#pragma unroll
        for (int kk = 0; kk < 32; kk += 4) {
            // A fragment (16x4): half-wave h holds K = kk+2h+{0,1}, M = lm
            v2f a, b;
            a.x = As[(kk + 2 * half + 0) * 16 + lm];
            a.y = As[(kk + 2 * half + 1) * 16 + lm];
            // B fragment (4x16): VGPR0 K=2h, VGPR1 K=2h+1, N = cb per half
            b.x = Bs[(kk + 2 * half + 0) * 64 + cb];
            b.y = Bs[(kk + 2 * half + 1) * 64 + cb];
            acc = __builtin_amdgcn_wmma_f32_16x16x4_f32(
                false, a, false, b, (short)0, acc, false, false);
        }
        __syncthreads();
    }

    // C/D layout: VGPR r -> M = m0 + r + 8*half, N = col
    if ((NOUT & 15) == 0 || col < NOUT) {
#pragma unroll
        for (int r = 0; r < 8; ++r) {
            Cout[(size_t)(m0 + r + 8 * half) * NOUT + col] = acc[r];
        }
    }
}

// ---------------------------------------------------------------------------
// Fused gather + normalize + scatter-add. Virtual edges e in [E, E+N) are
// self loops (src = dst = e - E), matching the reference's concat of loops.
// CPE = 4-float chunks per edge (power of two; chunks >= F/4 masked off).
// ---------------------------------------------------------------------------
template <int F, int CPE, int LSH>
__global__ void gcn_scatter(const float* __restrict__ feat,
                            const int* __restrict__ row,
                            const int* __restrict__ col,
                            const float* __restrict__ dinv,
                            float* __restrict__ out, int nE, int nNodes) {
    const int total = (nE + nNodes) * CPE;   // <= 211M, fits int32
    for (int i = blockIdx.x * blockDim.x + threadIdx.x; i < total;
         i += gridDim.x * blockDim.x) {
        const int e = i >> LSH;
        const int c = (i & (CPE - 1)) << 2;
        if (c >= F) continue;
        int s, t;
        if (e < nE) {
            s = row[e];
            t = col[e];
        } else {
            s = t = e - nE;
        }
        const float nm = dinv[s] * dinv[t];
        const float4 v = *(const float4*)(feat + (size_t)s * F + c);
        float* o = out + (size_t)t * F + c;
        atomicAdd(o + 0, v.x * nm);
        atomicAdd(o + 1, v.y * nm);
        atomicAdd(o + 2, v.z * nm);
        atomicAdd(o + 3, v.w * nm);
    }
}

// ---------------------------------------------------------------------------
// h = relu(h + b1)  (in place, H = 256 so bias index = i & 255)
// ---------------------------------------------------------------------------
__global__ void gcn_bias_relu(float* __restrict__ h, const float* __restrict__ b,
                              int total) {
    int i = blockIdx.x * blockDim.x + threadIdx.x;
    if (i < total) {
        float v = h[i] + b[i & (HID - 1)];
        h[i] = fmaxf(v, 0.0f);
    }
}

// ---------------------------------------------------------------------------
// out = log_softmax(agg2 + b2) over 40 classes; one wave32 per node, lane
// holds classes {lane, lane+32}; wave reduction via __shfl_xor.
// ---------------------------------------------------------------------------
__global__ void gcn_logsoftmax(const float* __restrict__ agg,
                               const float* __restrict__ b2,
                               float* __restrict__ out, int nNodes) {
    const int wavesPerBlock = blockDim.x >> 5;
    const int node = blockIdx.x * wavesPerBlock + (threadIdx.x >> 5);
    if (node >= nNodes) return;
    const int lane = threadIdx.x & 31;

    const float* a = agg + (size_t)node * NCLS;
    float v0 = a[lane] + b2[lane];                 // lane 0..31 all < 40
    float v1 = -INFINITY;
    if (lane < NCLS - 32) v1 = a[lane + 32] + b2[lane + 32];

    float m = fmaxf(v0, v1);
#pragma unroll
    for (int off = 16; off; off >>= 1) m = fmaxf(m, __shfl_xor(m, off, 32));

    float s = __expf(v0 - m) + ((lane < NCLS - 32) ? __expf(v1 - m) : 0.0f);
#pragma unroll
    for (int off = 16; off; off >>= 1) s += __shfl_xor(s, off, 32);

    const float lse = m + __logf(s);
    float* o = out + (size_t)node * NCLS;
    o[lane] = v0 - lse;
    if (lane < NCLS - 32) o[lane + 32] = v1 - lse;
}

// ---------------------------------------------------------------------------
extern "C" void kernel_launch(void* const* d_in, const int* in_sizes, int n_in,
                              void* d_out, int out_size, void* d_ws, size_t ws_size,
                              hipStream_t stream) {
    const float* x  = (const float*)d_in[0];   // [N, 512]
    const int*   ei = (const int*)d_in[1];     // [2, E] (int32 after jax canonicalization)
    const float* W1 = (const float*)d_in[2];   // [512, 256]
    const float* b1 = (const float*)d_in[3];   // [256]
    const float* W2 = (const float*)d_in[4];   // [256, 40]
    const float* b2 = (const float*)d_in[5];   // [40]
    const int E = in_sizes[1] / 2;
    const int n = NNODES;

    const int* row = ei;
    const int* col = ei + E;

    // Workspace layout (floats): dinv | xw1 | h(agg1) | hw2 | agg2
    float* ws   = (float*)d_ws;
    float* dinv = ws;                               // n (deg accum -> dinv)
    float* xw1  = ws + (1 << 17);                   // n*256
    float* h    = xw1 + (size_t)n * HID;            // n*256 (agg1, relu'd in place)
    float* hw2  = h + (size_t)n * HID;              // n*40
    float* agg2 = hw2 + (size_t)n * NCLS;           // n*40

    hipMemsetAsync(dinv, 0, (size_t)n * sizeof(float), stream);
    hipMemsetAsync(h, 0, (size_t)n * HID * sizeof(float), stream);
    hipMemsetAsync(agg2, 0, (size_t)n * NCLS * sizeof(float), stream);

    gcn_degree<<<1024, 256, 0, stream>>>(col, dinv, E);
    gcn_rsqrt<<<(n + 255) / 256, 256, 0, stream>>>(dinv, n);

    // Layer 1: xw1 = x @ W1, aggregate, bias+relu
    dim3 g1(n / 16, HID / 64);                      // 6250 x 4
    gcn_gemm_wmma<FIN, HID><<<g1, 128, 0, stream>>>(x, W1, xw1);
    gcn_scatter<HID, 64, 6><<<2048, 256, 0, stream>>>(xw1, row, col, dinv, h, E, n);
    gcn_bias_relu<<<(n * HID + 255) / 256, 256, 0, stream>>>(h, b1, n * HID);

    // Layer 2: hw2 = h @ W2, aggregate, log_softmax
    dim3 g2(n / 16, 1);
    gcn_gemm_wmma<HID, NCLS><<<g2, 128, 0, stream>>>(h, W2, hw2);
    gcn_scatter<NCLS, 16, 4><<<2048, 256, 0, stream>>>(hw2, row, col, dinv, agg2, E, n);
    gcn_logsoftmax<<<(n + 7) / 8, 256, 0, stream>>>(agg2, b2, (float*)d_out, n);
}